// LRRUCSPN_75935021793729
// MI455X (gfx1250) — compile-verified
//
#include <hip/hip_runtime.h>

// LRRU/CSPN deformable 3x3 refinement, B=4, H=352, W=1216, K=3, PAD=1, DIL=1.
// Strategy: stage depth into a zero-padded image P in d_ws, then bilinear
// sampling needs ZERO bounds logic: clamp float coords to [-1,H]x[-1,W]
// (exact per zero-outside semantics), gather pairs with one b64 per row,
// second row via immediate offset. Streams use NT hints; depth/P stay in L2.

typedef float v2f    __attribute__((ext_vector_type(2)));
typedef float v2f_a4 __attribute__((ext_vector_type(2), aligned(4)));

#define Kc   3
#define KK   9
#define PADc 1
#define Bv   4
#define Hv   352
#define Wv   1216
#define HWv  (Hv * Wv)

#define Wp   1220          // padded row stride (>= Wv+3, 16B-multiple-ish)
#define Hp   (Hv + 3)      // rows 0..Hv+2 ; zero rows at 0, Hv+1, Hv+2
#define PHWv (Hp * Wp)     // per-batch padded size (floats)

// ---------------- pad kernel: P[b][1+y][1+x] = depth, zero border ----------
__global__ __launch_bounds__(256) void lrru_pad_kernel(
    const float* __restrict__ depth, float* __restrict__ P)
{
    const int xx = blockIdx.x * 256 + threadIdx.x;
    const int yy = blockIdx.y;
    const int b  = blockIdx.z;
    if (xx >= Wp) return;
    const int y = yy - 1;
    const int x = xx - 1;
    const bool valid = ((unsigned)y < (unsigned)Hv) && ((unsigned)x < (unsigned)Wv);
    const float v = valid ? depth[(size_t)b * HWv + y * Wv + x] : 0.0f;
    P[(size_t)b * PHWv + yy * Wp + xx] = v;
}

// ---------------- padded bilinear: no bounds logic at all ------------------
__device__ __forceinline__ float bil_padded(const float* __restrict__ P,
                                            float y, float x)
{
    // y,x are already in padded coords (pad shift cancels -PAD).
    const float yc = fminf(fmaxf(y, 0.0f), (float)(Hv + 1));
    const float xc = fminf(fmaxf(x, 0.0f), (float)(Wv + 1));
    const float y0f = floorf(yc);
    const float x0f = floorf(xc);
    const float fy = yc - y0f;
    const float fx = xc - x0f;
    const unsigned iy = (unsigned)y0f;
    const unsigned ix = (unsigned)x0f;

    const float* r = P + (iy * (unsigned)Wp + ix);
    const v2f_a4 pa = *(const v2f_a4*)(r);        // row iy   : cols ix, ix+1
    const v2f_a4 pb = *(const v2f_a4*)(r + Wp);   // row iy+1 : imm offset +4880B

    const float row0 = fmaf(fx, pa.y - pa.x, pa.x);
    const float row1 = fmaf(fx, pb.y - pb.x, pb.x);
    return fmaf(fy, row1 - row0, row0);
}

__global__ __launch_bounds__(128) void lrru_cspn_padded_kernel(
    const float* __restrict__ depth,    // [B,1,H,W] (residual only)
    const float* __restrict__ weight,   // [B,9,H,W]
    const float* __restrict__ offset,   // [B,18,H,W]
    const float* __restrict__ w_param,  // [1,1,3,3]
    const float* __restrict__ b_param,  // [1]
    const float* __restrict__ P,        // [B,Hp,Wp] padded depth
    float* __restrict__ out)            // [B,1,H,W]
{
    const int w0 = (blockIdx.x * 128 + threadIdx.x) * 2;   // 2 px / thread
    const int h  = blockIdx.y;
    const int b  = blockIdx.z;
    if (w0 >= Wv) return;

    const int p   = h * Wv + w0;
    const int HW2 = HWv / 2;

    const float* __restrict__ dimg = depth + (size_t)b * HWv;
    const float* __restrict__ Pb   = P     + (size_t)b * PHWv;
    const v2f*   __restrict__ wgt  = (const v2f*)(weight + (size_t)b * KK * HWv + p);
    const v2f*   __restrict__ off  = (const v2f*)(offset + (size_t)b * 2 * KK * HWv + p);

    // Prefetch the 3-row padded neighborhood (gfx1250 global_prefetch_b8).
    __builtin_prefetch(Pb + (size_t)(h    ) * Wp + w0, 0, 3);
    __builtin_prefetch(Pb + (size_t)(h + 1) * Wp + w0, 0, 3);
    __builtin_prefetch(Pb + (size_t)(h + 2) * Wp + w0, 0, 3);

    // 9 mask weights for both pixels (single-use stream -> non-temporal).
    v2f wk[KK];
    float sx = 0.0f, sy = 0.0f;
    #pragma unroll
    for (int k = 0; k < KK; ++k) {
        wk[k] = __builtin_nontemporal_load(wgt + k * HW2);
        sx += wk[k].x;
        sy += wk[k].y;
    }
    const float meanx = sx * (1.0f / 9.0f);
    const float meany = sy * (1.0f / 9.0f);

    float tap[KK];
    #pragma unroll
    for (int k = 0; k < KK; ++k) tap[k] = w_param[k];
    const float bias = b_param[0];

    float accx = 0.0f, accy = 0.0f;
    #pragma unroll
    for (int k = 0; k < KK; ++k) {
        const v2f dy = __builtin_nontemporal_load(off + (2 * k)     * HW2);
        const v2f dx = __builtin_nontemporal_load(off + (2 * k + 1) * HW2);

        // padded base coords: (+1 pad) cancels (-PAD)
        const float by  = (float)(h  + (k / Kc));
        const float bx0 = (float)(w0 + (k % Kc));

        const float s0 = bil_padded(Pb, by + dy.x, bx0 + dx.x);
        const float s1 = bil_padded(Pb, by + dy.y, bx0 + 1.0f + dx.y);

        accx = fmaf(s0 * (wk[k].x - meanx), tap[k], accx);
        accy = fmaf(s1 * (wk[k].y - meany), tap[k], accy);
    }

    const v2f ctr = *(const v2f*)(dimg + p);
    v2f res;
    res.x = accx + bias + ctr.x;
    res.y = accy + bias + ctr.y;
    __builtin_nontemporal_store(res, (v2f*)(out + (size_t)b * HWv + p));
}

// ---------------- fallback (no workspace): round-3 direct version ----------
__device__ __forceinline__ float bilinear_zero_pad(const float* __restrict__ img,
                                                   float y, float x)
{
    const float y0f = floorf(y);
    const float x0f = floorf(x);
    const float wy1r = y - y0f;
    const float wx1  = x - x0f;
    const float wy0r = 1.0f - wy1r;
    const float wx0  = 1.0f - wx1;

    const int iy0 = (int)y0f;
    const int ix0 = (int)x0f;
    const int iy1 = iy0 + 1;

    const float wy0 = ((unsigned)iy0 < (unsigned)Hv) ? wy0r : 0.0f;
    const float wy1 = ((unsigned)iy1 < (unsigned)Hv) ? wy1r : 0.0f;

    const bool in_  = ((unsigned)ix0 < (unsigned)(Wv - 1));
    const bool led  = (ix0 == -1);
    const bool red  = (ix0 == Wv - 1);
    const float ax = in_ ? wx0 : (led ? wx1 : 0.0f);
    const float ay = in_ ? wx1 : (red ? wx0 : 0.0f);

    const unsigned cy0 = (unsigned)min(max(iy0, 0), Hv - 1);
    const unsigned cy1 = (unsigned)min(max(iy1, 0), Hv - 1);
    const unsigned cx  = (unsigned)min(max(ix0, 0), Wv - 2);

    const v2f_a4 pa = *(const v2f_a4*)(img + (cy0 * (unsigned)Wv + cx));
    const v2f_a4 pb = *(const v2f_a4*)(img + (cy1 * (unsigned)Wv + cx));

    const float row0 = fmaf(ax, pa.x, ay * pa.y);
    const float row1 = fmaf(ax, pb.x, ay * pb.y);
    return fmaf(wy0, row0, wy1 * row1);
}

__global__ __launch_bounds__(128) void lrru_cspn_direct_kernel(
    const float* __restrict__ depth,
    const float* __restrict__ weight,
    const float* __restrict__ offset,
    const float* __restrict__ w_param,
    const float* __restrict__ b_param,
    float* __restrict__ out)
{
    const int w0 = (blockIdx.x * 128 + threadIdx.x) * 2;
    const int h  = blockIdx.y;
    const int b  = blockIdx.z;
    if (w0 >= Wv) return;

    const int p   = h * Wv + w0;
    const int HW2 = HWv / 2;

    const float* __restrict__ dimg = depth + (size_t)b * HWv;
    const v2f*   __restrict__ wgt  = (const v2f*)(weight + (size_t)b * KK * HWv + p);
    const v2f*   __restrict__ off  = (const v2f*)(offset + (size_t)b * 2 * KK * HWv + p);

    v2f wk[KK];
    float sx = 0.0f, sy = 0.0f;
    #pragma unroll
    for (int k = 0; k < KK; ++k) {
        wk[k] = __builtin_nontemporal_load(wgt + k * HW2);
        sx += wk[k].x;
        sy += wk[k].y;
    }
    const float meanx = sx * (1.0f / 9.0f);
    const float meany = sy * (1.0f / 9.0f);

    float tap[KK];
    #pragma unroll
    for (int k = 0; k < KK; ++k) tap[k] = w_param[k];
    const float bias = b_param[0];

    float accx = 0.0f, accy = 0.0f;
    #pragma unroll
    for (int k = 0; k < KK; ++k) {
        const v2f dy = __builtin_nontemporal_load(off + (2 * k)     * HW2);
        const v2f dx = __builtin_nontemporal_load(off + (2 * k + 1) * HW2);
        const float by  = (float)(h  + (k / Kc) - PADc);
        const float bx0 = (float)(w0 + (k % Kc) - PADc);
        const float s0 = bilinear_zero_pad(dimg, by + dy.x, bx0 + dx.x);
        const float s1 = bilinear_zero_pad(dimg, by + dy.y, bx0 + 1.0f + dx.y);
        accx = fmaf(s0 * (wk[k].x - meanx), tap[k], accx);
        accy = fmaf(s1 * (wk[k].y - meany), tap[k], accy);
    }

    const v2f ctr = *(const v2f*)(dimg + p);
    v2f res;
    res.x = accx + bias + ctr.x;
    res.y = accy + bias + ctr.y;
    __builtin_nontemporal_store(res, (v2f*)(out + (size_t)b * HWv + p));
}

extern "C" void kernel_launch(void* const* d_in, const int* in_sizes, int n_in,
                              void* d_out, int out_size, void* d_ws, size_t ws_size,
                              hipStream_t stream) {
    (void)in_sizes; (void)n_in; (void)out_size;
    const float* depth   = (const float*)d_in[0];
    const float* weight  = (const float*)d_in[1];
    const float* offset  = (const float*)d_in[2];
    const float* w_param = (const float*)d_in[3];
    const float* b_param = (const float*)d_in[4];
    float* out = (float*)d_out;

    const size_t ws_needed = (size_t)Bv * PHWv * sizeof(float);   // ~6.93 MB

    dim3 blockM(128, 1, 1);
    dim3 gridM((Wv / 2 + 127) / 128, Hv, Bv);   // 5 x 352 x 4

    if (d_ws != nullptr && ws_size >= ws_needed) {
        float* P = (float*)d_ws;
        dim3 blockP(256, 1, 1);
        dim3 gridP((Wp + 255) / 256, Hp, Bv);    // 5 x 355 x 4
        lrru_pad_kernel<<<gridP, blockP, 0, stream>>>(depth, P);
        lrru_cspn_padded_kernel<<<gridM, blockM, 0, stream>>>(
            depth, weight, offset, w_param, b_param, P, out);
    } else {
        lrru_cspn_direct_kernel<<<gridM, blockM, 0, stream>>>(
            depth, weight, offset, w_param, b_param, out);
    }
}